// ContinuousLocationMap_27960237097539
// MI455X (gfx1250) — compile-verified
//
#include <hip/hip_runtime.h>

// ContinuousLocationMap: out[B,256,256,4] f32.
//  channels 0,1 = 1.0 at hit bins; channels 2,3 = raw (x,y) at hit bins; zero elsewhere.
// Phase 1: zero-fill 256MB output via CDNA5 async LDS->global b128 stores (ASYNCcnt
//          path), 8 stores unrolled with immediate offsets, non-temporal hint.
// Phase 2: scatter 32768 points, one aligned 16B store each.

#define BINS_XY 256

// Compile-time block geometry (launch must match NT).
constexpr unsigned NT     = 256;         // threads per block
constexpr unsigned UNROLL = 8;           // async stores per chunk iteration
constexpr unsigned CHUNK  = NT * UNROLL; // float4 per block-chunk (2048 = 32KB)

__global__ void clm_zero_fill_async(float4* __restrict__ out, unsigned n4) {
  // 4KB of zeros in LDS; each lane streams its own 16B slice to global via the
  // async pipe (global_store_async_from_lds_b128, tracked by ASYNCcnt).
  __shared__ __attribute__((aligned(16))) float zbuf[NT * 4];
  const unsigned t = threadIdx.x;
  float4 z;
  z.x = 0.0f; z.y = 0.0f; z.z = 0.0f; z.w = 0.0f;
  *reinterpret_cast<float4*>(&zbuf[t * 4]) = z;
  __syncthreads();

  // Low 32 bits of the flat (generic) address of an LDS object are the LDS byte
  // offset (ISA 10.2).
  const unsigned lds_off = (unsigned)(size_t)(&zbuf[t * 4]);

  const unsigned nchunk = n4 / CHUNK;  // constant divisor -> shift

  // Main loop: one address computation per 8 stores; immediate offsets k*4096B.
  // Each unrolled store is a contiguous 4KB write per block (fully coalesced).
  // NT store hint: 256MB streaming fill exceeds the 192MB L2 -> don't thrash it.
  for (unsigned c = blockIdx.x; c < nchunk; c += gridDim.x) {
    const float4* p = out + ((size_t)c * CHUNK + t);
    asm volatile(
        "global_store_async_from_lds_b128 %0, %1, off th:TH_STORE_NT\n"
        "global_store_async_from_lds_b128 %0, %1, off offset:4096 th:TH_STORE_NT\n"
        "global_store_async_from_lds_b128 %0, %1, off offset:8192 th:TH_STORE_NT\n"
        "global_store_async_from_lds_b128 %0, %1, off offset:12288 th:TH_STORE_NT\n"
        "global_store_async_from_lds_b128 %0, %1, off offset:16384 th:TH_STORE_NT\n"
        "global_store_async_from_lds_b128 %0, %1, off offset:20480 th:TH_STORE_NT\n"
        "global_store_async_from_lds_b128 %0, %1, off offset:24576 th:TH_STORE_NT\n"
        "global_store_async_from_lds_b128 %0, %1, off offset:28672 th:TH_STORE_NT\n"
        :
        : "v"(p), "v"(lds_off)
        : "memory");
  }

  // Remainder (empty for the actual 16M-float4 size): plain wide stores.
  for (unsigned i = nchunk * CHUNK + blockIdx.x * NT + t; i < n4;
       i += gridDim.x * NT) {
    out[i] = z;
  }

  // Drain the async pipe (also implicit at s_endpgm, but make it explicit).
  asm volatile("s_wait_asynccnt 0x0" ::: "memory");
}

__global__ void clm_scatter(const float* __restrict__ batch,
                            float* __restrict__ out,
                            int n_points, int L) {
  const int i = blockIdx.x * blockDim.x + threadIdx.x;
  if (i >= n_points) return;

  const float x = batch[2 * i + 0];
  const float y = batch[2 * i + 1];

  // loc_delta = (10 - 0) / 256, computed exactly as the reference does; index is
  // trunc-toward-zero of (v/delta + 0.5) (values are non-negative).
  const float delta = (10.0f - 0.0f) / (float)BINS_XY;
  int ix = (int)(x / delta + 0.5f);
  int iy = (int)(y / delta + 0.5f);
  ix = min(max(ix, 0), BINS_XY - 1);
  iy = min(max(iy, 0), BINS_XY - 1);

  const int b = i / L;
  const size_t off = ((((size_t)b * BINS_XY) + (size_t)ix) * BINS_XY + (size_t)iy) * 4;

  float4 v;
  v.x = 1.0f;  // corr channel 0
  v.y = 1.0f;  // corr channel 1
  v.z = x;     // loc x
  v.w = y;     // loc y
  *reinterpret_cast<float4*>(out + off) = v;  // one aligned global_store_b128
}

extern "C" void kernel_launch(void* const* d_in, const int* in_sizes, int n_in,
                              void* d_out, int out_size, void* d_ws, size_t ws_size,
                              hipStream_t stream) {
  (void)n_in; (void)d_ws; (void)ws_size;

  const float* batch = (const float*)d_in[0];
  float* out = (float*)d_out;

  const int n_points = in_sizes[0] / 2;                       // B*L
  const int B = out_size / (BINS_XY * BINS_XY * 4);           // 256
  const int L = (B > 0) ? (n_points / B) : 1;                 // 128
  const unsigned n4 = (unsigned)(out_size / 4);               // 16M float4 stores

  // Phase 1: zero fill. 2048 blocks x NT threads; 4 chunk-iterations each,
  // 8 async b128 stores per iteration.
  clm_zero_fill_async<<<dim3(2048), dim3(NT), 0, stream>>>((float4*)out, n4);

  // Phase 2: scatter (stream-ordered after the fill).
  const int blocks = (n_points + NT - 1) / NT;
  clm_scatter<<<dim3(blocks), dim3(NT), 0, stream>>>(batch, out, n_points, L);
}